// DiscriminativeLoss_30477087933069
// MI455X (gfx1250) — compile-verified
//
#include <hip/hip_runtime.h>
#include <hip/hip_bf16.h>
#include <math.h>

typedef float v2f __attribute__((ext_vector_type(2)));
typedef float v8f __attribute__((ext_vector_type(8)));

#define BATCH 4
#define DIM 16
#define KINST 12
#define KPAD 16
#define HWSZ (512*512)
#define TILE 256
#define BLOCKS_PER_B 256
#define TILES_PER_BLOCK ((HWSZ / TILE) / BLOCKS_PER_B)  // = 4
#define RSTRIDE 260   // dwords; rows 16B aligned, 260%64==4 -> conflict-free WMMA operand reads

// workspace layout (floats)
#define WS_SUMS   0       // [B][16][16]  sums[b][inst][d]
#define WS_COUNTS 1024    // [B][16]
#define WS_PULL   1088    // [B][16]
#define WS_MEANST 1152    // [B][16][16]  meansT[b][d][inst]
#define WS_MSQ    2176    // [B][16]
#define WS_ZERO_N 1152

// ---- CDNA5 async global->LDS (ASYNCcnt-tracked), 16B per lane per instruction ----
__device__ __forceinline__ void async_copy16(unsigned int lds_byte_off, const void* gaddr) {
    asm volatile("global_load_async_to_lds_b128 %0, %1, off"
                 :: "v"(lds_byte_off), "v"((unsigned long long)(size_t)gaddr)
                 : "memory");
}
#define WAIT_ASYNC_0() asm volatile("s_wait_asynccnt 0x0" ::: "memory")

__global__ void zero_ws_kernel(float* __restrict__ ws) {
    int i = blockIdx.x * blockDim.x + threadIdx.x;
    if (i < WS_ZERO_N) ws[i] = 0.0f;
}

// Pass 1: sums[b,k,d] = sum_hw mask[b,k,hw] * emb[b,d,hw]  via V_WMMA_F32_16X16X4_F32
// A = mask (M=inst x K=4 pixels), B = emb (K=4 pixels x N=d), C accumulates across tiles.
__global__ void k1_sums(const float* __restrict__ emb, const int* __restrict__ msk,
                        float* __restrict__ ws) {
    __shared__ __attribute__((aligned(16))) float ldsE[2][DIM][RSTRIDE];
    __shared__ __attribute__((aligned(16))) int   ldsM[2][KINST + 1][RSTRIDE]; // row 12 = zero pad
    __shared__ float ldsRed[KPAD * DIM];
    __shared__ float ldsCnt[KPAD];

    const int tid  = threadIdx.x;
    const int lane = tid & 31;
    const int wave = tid >> 5;
    const int b    = blockIdx.x / BLOCKS_PER_B;
    const int blk  = blockIdx.x % BLOCKS_PER_B;
    const float* embB = emb + (size_t)b * DIM * HWSZ;
    const int*   mskB = msk + (size_t)b * KINST * HWSZ;

    const int nidx = lane & 15;                         // inst for A / d for B
    const int koff = (lane >> 4) * 2;                   // K-slot base per lane-half
    const int mrow = (nidx < KINST) ? nidx : KINST;     // clamp -> zero row (branch-free)

    auto issue = [&](int t, int s) {
        const int p0 = (blk + t * BLOCKS_PER_B) * TILE;
#pragma unroll
        for (int i = 0; i < 4; ++i) {
            int c = i * 256 + tid;
            int d = c >> 6;
            int px = (c & 63) << 2;
            async_copy16((unsigned int)(size_t)&ldsE[s][d][px],
                         embB + (size_t)d * HWSZ + p0 + px);
        }
#pragma unroll
        for (int i = 0; i < 3; ++i) {
            int c = i * 256 + tid;
            int k = c >> 6;
            int px = (c & 63) << 2;
            async_copy16((unsigned int)(size_t)&ldsM[s][k][px],
                         mskB + (size_t)k * HWSZ + p0 + px);
        }
    };

    // zero the pad row of both buffers once (visible after first barrier)
    for (int i = tid; i < RSTRIDE; i += 256) {
        ldsM[0][KINST][i] = 0;
        ldsM[1][KINST][i] = 0;
    }

    v8f acc = {};
    float cnt[KINST];
#pragma unroll
    for (int k = 0; k < KINST; ++k) cnt[k] = 0.0f;

    issue(0, 0);
    for (int t = 0; t < TILES_PER_BLOCK; ++t) {
        const int s = t & 1;
        WAIT_ASYNC_0();        // tile t landed in LDS
        __syncthreads();       // visible to all waves; prior buffer free
        if (t + 1 < TILES_PER_BLOCK) issue(t + 1, s ^ 1);  // overlap with compute

        // counts from LDS (coalesced, conflict-free)
#pragma unroll
        for (int k = 0; k < KINST; ++k) cnt[k] += (float)ldsM[s][k][tid];

        // each wave: 32 pixels -> 8 WMMAs of 4 pixels
        const int wp = wave * 32;
#pragma unroll
        for (int c4 = 0; c4 < 8; ++c4) {
            const int po = wp + c4 * 4 + koff;           // even -> 8B aligned
            int2 mv = *(const int2*)&ldsM[s][mrow][po];  // ds_load_b64, no branches
            v2f  bv = *(const v2f*)&ldsE[s][nidx][po];   // ds_load_b64
            v2f  a;
            a.x = (float)mv.x;
            a.y = (float)mv.y;
            acc = __builtin_amdgcn_wmma_f32_16x16x4_f32(
                false, a, false, bv, (short)0, acc, false, false);
        }
        __syncthreads();       // all waves done reading buffer s before next overwrite cycle
    }

    // block reduce, then one global atomicAdd per element
    if (tid < KPAD * DIM) ldsRed[tid] = 0.0f;
    if (tid < KPAD) ldsCnt[tid] = 0.0f;
    __syncthreads();
    const int mhi = (lane >> 4) * 8;
#pragma unroll
    for (int i = 0; i < 8; ++i)
        atomicAdd(&ldsRed[(i + mhi) * DIM + nidx], acc[i]);
#pragma unroll
    for (int k = 0; k < KINST; ++k)
        atomicAdd(&ldsCnt[k], cnt[k]);
    __syncthreads();
    if (tid < KPAD * DIM) atomicAdd(&ws[WS_SUMS + b * KPAD * DIM + tid], ldsRed[tid]);
    if (tid < KPAD)       atomicAdd(&ws[WS_COUNTS + b * KPAD + tid], ldsCnt[tid]);
}

// Pass 2 (tiny): means^T and mean_sq
__global__ void k2_means(float* __restrict__ ws) {
    const int tid = threadIdx.x;
    for (int idx = tid; idx < BATCH * KPAD * DIM; idx += blockDim.x) {
        int b = idx >> 8, k = (idx >> 4) & 15, d = idx & 15;
        float c  = ws[WS_COUNTS + b * KPAD + k];
        float sc = fmaxf(c, 1.0f);
        ws[WS_MEANST + b * 256 + d * 16 + k] = ws[WS_SUMS + b * 256 + k * 16 + d] / sc;
    }
    for (int idx = tid; idx < BATCH * KPAD; idx += blockDim.x) {
        int b = idx >> 4, k = idx & 15;
        float c  = ws[WS_COUNTS + b * KPAD + k];
        float sc = fmaxf(c, 1.0f);
        float s = 0.0f;
        for (int d = 0; d < DIM; ++d) {
            float mean = ws[WS_SUMS + b * 256 + k * 16 + d] / sc;
            s += mean * mean;
        }
        ws[WS_MSQ + idx] = s;
    }
}

// Pass 3: dot[b,k,hw] via 4 chained WMMAs over d, hinge^2 masked accumulate -> pull_sum[b,k]
__global__ void k3_pull(const float* __restrict__ emb, const int* __restrict__ msk,
                        float* __restrict__ ws) {
    __shared__ __attribute__((aligned(16))) float ldsE[2][DIM][RSTRIDE];
    __shared__ __attribute__((aligned(16))) int   ldsM[2][KINST + 1][RSTRIDE]; // row 12 = zero pad
    __shared__ __attribute__((aligned(16))) float ldsSq[TILE];
    __shared__ float ldsP[KPAD];

    const int tid  = threadIdx.x;
    const int lane = tid & 31;
    const int wave = tid >> 5;
    const int b    = blockIdx.x / BLOCKS_PER_B;
    const int blk  = blockIdx.x % BLOCKS_PER_B;
    const float* embB = emb + (size_t)b * DIM * HWSZ;
    const int*   mskB = msk + (size_t)b * KINST * HWSZ;

    const int nidx = lane & 15;
    const int koff = (lane >> 4) * 2;
    const int mrow = (nidx < KINST) ? nidx : KINST;

    auto issue = [&](int t, int s) {
        const int p0 = (blk + t * BLOCKS_PER_B) * TILE;
#pragma unroll
        for (int i = 0; i < 4; ++i) {
            int c = i * 256 + tid;
            int d = c >> 6;
            int px = (c & 63) << 2;
            async_copy16((unsigned int)(size_t)&ldsE[s][d][px],
                         embB + (size_t)d * HWSZ + p0 + px);
        }
#pragma unroll
        for (int i = 0; i < 3; ++i) {
            int c = i * 256 + tid;
            int k = c >> 6;
            int px = (c & 63) << 2;
            async_copy16((unsigned int)(size_t)&ldsM[s][k][px],
                         mskB + (size_t)k * HWSZ + p0 + px);
        }
    };

    for (int i = tid; i < RSTRIDE; i += 256) {
        ldsM[0][KINST][i] = 0;
        ldsM[1][KINST][i] = 0;
    }

    // preload B operand (means^T) and this lane's mean_sq
    v2f bop[4];
#pragma unroll
    for (int j = 0; j < 4; ++j) {
        const int d0 = j * 4 + koff;
        bop[j].x = ws[WS_MEANST + b * 256 + d0 * 16 + nidx];
        bop[j].y = ws[WS_MEANST + b * 256 + (d0 + 1) * 16 + nidx];
    }
    const float msq = ws[WS_MSQ + b * KPAD + nidx];
    float pull_local = 0.0f;

    issue(0, 0);
    for (int t = 0; t < TILES_PER_BLOCK; ++t) {
        const int s = t & 1;
        WAIT_ASYNC_0();
        __syncthreads();
        if (t + 1 < TILES_PER_BLOCK) issue(t + 1, s ^ 1);

        // per-pixel ||e||^2 from LDS (coalesced, conflict-free)
        float sq = 0.0f;
#pragma unroll
        for (int d = 0; d < DIM; ++d) {
            float v = ldsE[s][d][tid];
            sq += v * v;
        }
        ldsSq[tid] = sq;
        __syncthreads();

#pragma unroll
        for (int g = 0; g < 2; ++g) {
            const int pg = wave * 32 + g * 16;
            v8f c = {};
#pragma unroll
            for (int j = 0; j < 4; ++j) {
                v2f a;
                const int p  = pg + nidx;        // A: M = pixel
                const int d0 = j * 4 + koff;     //    K = d slot
                a.x = ldsE[s][d0][p];
                a.y = ldsE[s][d0 + 1][p];
                c = __builtin_amdgcn_wmma_f32_16x16x4_f32(
                    false, a, false, bop[j], (short)0, c, false, false);
            }
            const int mhi  = (lane >> 4) * 8;
            const int base = pg + mhi;                        // 8-dword aligned
            int4   ma = *(const int4*)&ldsM[s][mrow][base];   // broadcast b128 loads
            int4   mb = *(const int4*)&ldsM[s][mrow][base + 4];
            float4 sa = *(const float4*)&ldsSq[base];
            float4 sb = *(const float4*)&ldsSq[base + 4];
            const float mvals[8] = {(float)ma.x, (float)ma.y, (float)ma.z, (float)ma.w,
                                    (float)mb.x, (float)mb.y, (float)mb.z, (float)mb.w};
            const float sqv[8]   = {sa.x, sa.y, sa.z, sa.w, sb.x, sb.y, sb.z, sb.w};
#pragma unroll
            for (int i = 0; i < 8; ++i) {
                float d2 = sqv[i] - 2.0f * c[i] + msq;
                d2 = fmaxf(d2, 0.0f);
                float dist  = sqrtf(d2);
                float hinge = fmaxf(dist - 0.5f, 0.0f);
                pull_local += hinge * hinge * mvals[i];
            }
        }
        __syncthreads();
    }

    if (tid < KPAD) ldsP[tid] = 0.0f;
    __syncthreads();
    atomicAdd(&ldsP[nidx], pull_local);
    __syncthreads();
    if (tid < KPAD) atomicAdd(&ws[WS_PULL + b * KPAD + tid], ldsP[tid]);
}

// Pass 4: scalar finalize (pull/push/reg)
__global__ void k4_final(const float* __restrict__ ws, float* __restrict__ out) {
    if (threadIdx.x != 0 || blockIdx.x != 0) return;
    float total = 0.0f;
    for (int b = 0; b < BATCH; ++b) {
        float counts[KINST], msq[KINST];
        bool  valid[KINST];
        float nv = 0.0f;
        for (int k = 0; k < KINST; ++k) {
            counts[k] = ws[WS_COUNTS + b * KPAD + k];
            msq[k]    = ws[WS_MSQ + b * KPAD + k];
            valid[k]  = counts[k] > 0.0f;
            if (valid[k]) nv += 1.0f;
        }
        float safe_nv = fmaxf(nv, 1.0f);

        float pull_img = 0.0f;
        for (int k = 0; k < KINST; ++k)
            if (valid[k]) pull_img += ws[WS_PULL + b * KPAD + k] / fmaxf(counts[k], 1.0f);
        pull_img = (nv > 0.0f) ? pull_img / safe_nv : 0.0f;

        float push_sum = 0.0f;
        for (int k = 0; k < KINST; ++k) {
            for (int l = k + 1; l < KINST; ++l) {
                if (valid[k] && valid[l]) {
                    float cross = 0.0f;
                    for (int d = 0; d < DIM; ++d)
                        cross += ws[WS_MEANST + b * 256 + d * 16 + k] *
                                 ws[WS_MEANST + b * 256 + d * 16 + l];
                    float pd2   = fmaxf(msq[k] + msq[l] - 2.0f * cross, 0.0f);
                    float pdist = sqrtf(pd2);
                    float h     = fmaxf(3.0f - pdist, 0.0f);  // 2*DELTA_PUSH
                    push_sum += h * h;
                }
            }
        }
        float npairs   = nv * (nv - 1.0f) * 0.5f;
        float push_img = (nv > 1.0f) ? push_sum / fmaxf(npairs, 1.0f) : 0.0f;

        float reg = 0.0f;
        for (int k = 0; k < KINST; ++k)
            if (valid[k]) reg += sqrtf(fmaxf(msq[k], 0.0f));
        reg = (nv > 0.0f) ? reg / safe_nv : 0.0f;

        total += pull_img + push_img + reg;
    }
    out[0] = total / (float)BATCH;
}

extern "C" void kernel_launch(void* const* d_in, const int* in_sizes, int n_in,
                              void* d_out, int out_size, void* d_ws, size_t ws_size,
                              hipStream_t stream) {
    const float* emb = (const float*)d_in[0];
    const int*   msk = (const int*)d_in[1];
    float* ws  = (float*)d_ws;
    float* out = (float*)d_out;

    hipLaunchKernelGGL(zero_ws_kernel, dim3((WS_ZERO_N + 255) / 256), dim3(256), 0, stream, ws);
    hipLaunchKernelGGL(k1_sums, dim3(BATCH * BLOCKS_PER_B), dim3(256), 0, stream, emb, msk, ws);
    hipLaunchKernelGGL(k2_means, dim3(1), dim3(256), 0, stream, ws);
    hipLaunchKernelGGL(k3_pull, dim3(BATCH * BLOCKS_PER_B), dim3(256), 0, stream, emb, msk, ws);
    hipLaunchKernelGGL(k4_final, dim3(1), dim3(1), 0, stream, ws, out);
}